// EmbeddingLayer_17454747091125
// MI455X (gfx1250) — compile-verified
//
#include <hip/hip_runtime.h>
#include <stdint.h>

typedef __attribute__((ext_vector_type(4))) float f4;

// Problem constants (match the reference)
constexpr int B   = 32;
constexpr int LC  = 2048;
constexpr int P   = 512;
constexpr int DC  = 128;
constexpr int DP  = 768;
constexpr int DOUT = DC + DP;          // 896
constexpr int NPOS = B * LC;           // 65536 char positions

constexpr int WAVES_PER_BLOCK = 8;     // 8 char positions per block
constexpr int THREADS = WAVES_PER_BLOCK * 32;  // 256

__global__ __launch_bounds__(THREADS, 1)
void embed_concat_kernel(const int*   __restrict__ chars,   // [B, LC]
                         const int*   __restrict__ offs,    // [B, P, 2] interleaved start,end
                         const float* __restrict__ pre,     // [B, P, DP]
                         const float* __restrict__ ce,      // [V, DC]
                         float*       __restrict__ out)     // [B, LC, DOUT]
{
    const int tid  = threadIdx.x;
    const int lane = tid & 31;
    const int wid  = tid >> 5;

    // First char position handled by this block; all 8 positions share one batch
    // item because LC (2048) is a multiple of WAVES_PER_BLOCK (8).
    const int block0 = blockIdx.x * WAVES_PER_BLOCK;
    const int b = block0 >> 11;                  // block0 / LC  (uniform, SGPR)

    // ---- Stage offsets[b] (P*2*4 = 4096 B) into LDS with async b128 copies ----
    // 256 threads x 16 B = 4096 B, one global_load_async_to_lds_b128 per lane.
    __shared__ int s_off[P * 2];
    {
        const int* gsrc = offs + b * (P * 2) + tid * 4;          // 16 B per thread
        unsigned   ldst = (unsigned)(uintptr_t)s_off + (unsigned)(tid * 16);
        asm volatile("global_load_async_to_lds_b128 %0, %1, off"
                     :: "v"(ldst), "v"(gsrc) : "memory");
    }

    // Wave-uniform char position
    int gw = __builtin_amdgcn_readfirstlane(block0 + wid);
    const int k = gw & (LC - 1);                 // position within sequence

    // Speculative WGP-scope prefetch of the *likely* pretrained row
    // (well-formed layout: subword p = k / SPAN). locality=3 -> nearest cache,
    // which on gfx1250 is a WGP-scope global_prefetch_b8 that fills all levels.
    __builtin_prefetch(pre + ((size_t)b * P + (size_t)(k >> 2)) * DP, 0, 3);

    // ---- Char-embedding gather (overlapped with the async LDS copy) ----
    const int token = chars[gw];                                  // scalar load
    const f4  cv    = ((const f4*)(ce + (size_t)token * DC))[lane];
    float* orow = out + (size_t)gw * DOUT;
    __builtin_nontemporal_store(cv, (f4*)orow + lane);            // 128 floats

    // ---- Wait for offsets table, then binary search in LDS ----
    asm volatile("s_wait_asynccnt 0" ::: "memory");
    __syncthreads();

    // searchsorted(starts, k, side='right') - 1, clipped to [0, P-1]
    int lo = 0, hi = P;
    while (lo < hi) {
        int mid = (lo + hi) >> 1;
        if (s_off[2 * mid] <= k) lo = mid + 1;
        else                     hi = mid;
    }
    int idx = lo - 1;
    if (idx < 0) idx = 0;                        // lo <= P so idx <= P-1 already
    idx = __builtin_amdgcn_readfirstlane(idx);

    const int st = __builtin_amdgcn_readfirstlane(s_off[2 * idx]);
    const int en = __builtin_amdgcn_readfirstlane(s_off[2 * idx + 1]);
    const bool valid = (k >= st) && (k < en);

    // ---- Subword part: 768 floats = 6 x b128 per lane, non-temporal out ----
    const f4* prow = (const f4*)(pre + ((size_t)b * P + idx) * DP);
    f4* osub = (f4*)(orow + DC);
    if (valid) {
#pragma unroll
        for (int j = 0; j < DP / (32 * 4); ++j) {                 // 6 iterations
            f4 v = prow[j * 32 + lane];
            __builtin_nontemporal_store(v, osub + j * 32 + lane);
        }
    } else {
        const f4 z = {0.0f, 0.0f, 0.0f, 0.0f};
#pragma unroll
        for (int j = 0; j < DP / (32 * 4); ++j)
            __builtin_nontemporal_store(z, osub + j * 32 + lane);
    }
}

extern "C" void kernel_launch(void* const* d_in, const int* in_sizes, int n_in,
                              void* d_out, int out_size, void* d_ws, size_t ws_size,
                              hipStream_t stream)
{
    const int*   chars = (const int*)  d_in[0];   // character_sequences [B, LC]
    const int*   offs  = (const int*)  d_in[1];   // offsets             [B, P, 2]
    const float* pre   = (const float*)d_in[2];   // pretrained          [B, P, DP]
    const float* ce    = (const float*)d_in[3];   // char_embedding      [V, DC]
    float*       out   = (float*)      d_out;     // [B, LC, DC+DP]

    const int blocks = NPOS / WAVES_PER_BLOCK;    // 8192
    embed_concat_kernel<<<blocks, THREADS, 0, stream>>>(chars, offs, pre, ce, out);
}